// Model_60713657697016
// MI455X (gfx1250) — compile-verified
//
#include <hip/hip_runtime.h>

// GroupNorm+Swish backward (dx only), fused single kernel for MI455X/gfx1250.
// N=16, C=256, H=W=64, G=32 -> Cg=8, hw=4096, npt=32768 per (n,g) group.
// Group data is contiguous: offset (n*G+g)*32768.
//
// Phase 1: async global->LDS DMA stream (ASYNCcnt pipeline), group reduction.
// Phase 2: L2-resident re-read (NT, last-use) -> recompute -> NT store of dx.
// HBM traffic ~= 192 MB (read x+dy once, write dx once) ~ 8.2 us @ 23.3 TB/s.

#define GROUPS   32
#define CG       8
#define HW       4096
#define NPT      (CG * HW)          // 32768 floats per group
#define TPB      256                // threads per block (8 waves of 32)
#define CHUNK    HW                 // one channel per chunk (gamma/beta uniform)
#define V4_PER_T (CHUNK / (TPB * 4))// 4 float4 per thread per chunk

typedef __attribute__((ext_vector_type(4))) float v4f;  // native vector for NT ops

// CDNA5 async global->LDS DMA (tracked by ASYNCcnt). GVS addressing:
// vdst = LDS byte address, vaddr = 32-bit byte offset, saddr = 64-bit base.
__device__ __forceinline__ void async_b128(unsigned lds_byte,
                                           const float* sbase,
                                           unsigned goff_bytes) {
  asm volatile("global_load_async_to_lds_b128 %0, %1, %2"
               :: "v"(lds_byte), "v"(goff_bytes), "s"(sbase)
               : "memory");
}

__device__ __forceinline__ float swish_dgn(float xn, float ga, float be,
                                           float yv) {
  float gn  = xn * ga + be;
  float den = __expf(-gn) + 1.0f;
  float dsw = (gn - gn / den + 1.0f) / den;
  return dsw * yv;
}

__global__ __launch_bounds__(TPB) void gn_swish_bwd(
    const float* __restrict__ dy,   const float* __restrict__ mean,
    const float* __restrict__ rstd, const float* __restrict__ x,
    const float* __restrict__ gamma,const float* __restrict__ beta,
    float* __restrict__ dx)
{
  // 2 tensors x 2 buffers x 16 KB = 64 KB static LDS (reused for reduction).
  __shared__ float xb[2][CHUNK];
  __shared__ float yb[2][CHUNK];

  const int    ng   = blockIdx.x;          // n*G + g
  const int    g    = ng & (GROUPS - 1);
  const int    t    = threadIdx.x;
  const size_t base = (size_t)ng * NPT;

  const float mu = mean[ng];
  const float rs = rstd[ng];

  const float* xg = x  + base;
  const float* yg = dy + base;

  float s1 = 0.0f, s2 = 0.0f;

  auto issue = [&](int c, int b) {
    #pragma unroll
    for (int k = 0; k < V4_PER_T; ++k) {
      const unsigned vec  = (unsigned)(t + k * TPB);            // float4 slot
      const unsigned goff = (unsigned)((c * CHUNK + vec * 4) * sizeof(float));
      const unsigned lx = (unsigned)(size_t)&xb[b][vec * 4];    // LDS offset
      const unsigned ly = (unsigned)(size_t)&yb[b][vec * 4];
      async_b128(lx, xg, goff);
      async_b128(ly, yg, goff);
    }
  };

  // ---- Phase 1: async-DMA stream through LDS, accumulate group sums ----
  issue(0, 0);
  #pragma unroll 1
  for (int c = 0; c < CG; ++c) {
    if (c + 1 < CG) {
      issue(c + 1, (c + 1) & 1);
      asm volatile("s_wait_asynccnt 0x8" ::: "memory"); // prev chunk landed
    } else {
      asm volatile("s_wait_asynccnt 0x0" ::: "memory");
    }
    const float ga = gamma[g * CG + c];
    const float be = beta [g * CG + c];
    const int   b  = c & 1;
    #pragma unroll
    for (int k = 0; k < V4_PER_T; ++k) {
      const int vec = t + k * TPB;
      #pragma unroll
      for (int i = 0; i < 4; ++i) {
        const float xv = xb[b][vec * 4 + i];
        const float yv = yb[b][vec * 4 + i];
        const float xn = (xv - mu) * rs;
        const float dgn = swish_dgn(xn, ga, be, yv);
        s1 += ga * dgn;
        s2 += ga * dgn * xn;
      }
    }
  }

  // ---- Reduction: wave32 shuffle tree, then cross-wave via reused LDS ----
  #pragma unroll
  for (int m = 16; m >= 1; m >>= 1) {
    s1 += __shfl_xor(s1, m, 32);
    s2 += __shfl_xor(s2, m, 32);
  }
  __syncthreads();                      // phase-1 LDS consumption done
  float* red1 = &xb[0][0];              // reuse buffers for 8-wave combine
  float* red2 = &xb[0][32];
  float* cval = &xb[0][64];
  const int wave = t >> 5, lane = t & 31;
  if (lane == 0) { red1[wave] = s1; red2[wave] = s2; }
  __syncthreads();
  if (t == 0) {
    float a = 0.0f, b2 = 0.0f;
    #pragma unroll
    for (int w = 0; w < TPB / 32; ++w) { a += red1[w]; b2 += red2[w]; }
    const float inv = 1.0f / (float)NPT;
    cval[0] = a  * inv;                 // c1
    cval[1] = b2 * inv;                 // c2
  }
  __syncthreads();
  const float c1 = cval[0];
  const float c2 = cval[1];

  // ---- Phase 2: last-use re-read (L2-resident) of x/dy, emit dx (NT) ----
  const v4f* x4 = (const v4f*)xg;
  const v4f* y4 = (const v4f*)yg;
  v4f*       o4 = (v4f*)(dx + base);
  #pragma unroll 1
  for (int c = 0; c < CG; ++c) {
    const float ga = gamma[g * CG + c];
    const float be = beta [g * CG + c];
    #pragma unroll
    for (int k = 0; k < V4_PER_T; ++k) {
      const int vec = c * (CHUNK / 4) + t + k * TPB;
      v4f xv = __builtin_nontemporal_load(&x4[vec]);  // last use -> NT
      v4f yv = __builtin_nontemporal_load(&y4[vec]);  // last use -> NT
      v4f ov;
      #pragma unroll
      for (int i = 0; i < 4; ++i) {
        const float xn = (xv[i] - mu) * rs;
        const float dgn = swish_dgn(xn, ga, be, yv[i]);
        ov[i] = rs * (dgn * ga - xn * c2 - c1);
      }
      __builtin_nontemporal_store(ov, &o4[vec]);      // write-once -> NT
    }
  }
}

extern "C" void kernel_launch(void* const* d_in, const int* in_sizes, int n_in,
                              void* d_out, int out_size, void* d_ws, size_t ws_size,
                              hipStream_t stream) {
  (void)n_in; (void)d_ws; (void)ws_size; (void)in_sizes;
  const float* dy    = (const float*)d_in[0];
  const float* mean  = (const float*)d_in[1];
  const float* rstd  = (const float*)d_in[2];
  const float* x     = (const float*)d_in[3];
  const float* gamma = (const float*)d_in[4];
  const float* beta  = (const float*)d_in[5];
  float*       out   = (float*)d_out;

  const int blocks = out_size / NPT;   // N*G = 512 for the reference shapes
  gn_swish_bwd<<<blocks, TPB, 0, stream>>>(dy, mean, rstd, x, gamma, beta, out);
}